// System_24970939859264
// MI455X (gfx1250) — compile-verified
//
#include <hip/hip_runtime.h>
#include <math.h>

// ---------------------------------------------------------------------------
// MI455X (gfx1250) observation kernel.
//
// Bandwidth-bound streaming problem: 4,194,304 rows x 7 f32 in, 7 f32 out.
// Traffic ~235 MB @ 23.3 TB/s => ~10 us floor; AI ~1.7 flop/B, so matrix ops
// are irrelevant — the CDNA5 feature that matters is the async global<->LDS
// DMA path (GLOBAL_LOAD_ASYNC_TO_LDS_B128 / GLOBAL_STORE_ASYNC_FROM_LDS_B128,
// ASYNCcnt + s_wait_asynccnt), giving fully coalesced 16B transfers with no
// VGPR round trip. Hot kernel is specialized tail-free (rows%256 handled by a
// tiny scalar tail kernel) so the copy loops unroll to 1 or 2 async loads
// and 1 or 2 async stores per thread (the second only for tid < 192).
// ---------------------------------------------------------------------------

#define ROWS_PER_BLOCK 256
#define FPR 7                                   // floats per row
#define ELEMS_PER_BLOCK (ROWS_PER_BLOCK * FPR)  // 1792 floats = 7168 B

// Async-LDS builtins take v4i32 pointers in explicit address spaces:
//   (v4i32 addrspace(1)* global, v4i32 addrspace(3)* lds, imm offset, imm cpol)
typedef int v4i __attribute__((ext_vector_type(4)));
typedef __attribute__((address_space(1))) v4i* gv4p;   // global v4i32*
typedef __attribute__((address_space(3))) v4i* lv4p;   // LDS    v4i32*

#if __has_builtin(__builtin_amdgcn_global_load_async_to_lds_b128) && \
    __has_builtin(__builtin_amdgcn_global_store_async_from_lds_b128) && \
    __has_builtin(__builtin_amdgcn_s_wait_asynccnt)
#define HAVE_ASYNC_LDS 1
#else
#define HAVE_ASYNC_LDS 0
#endif

// ---- per-row observation math (bit-exact mirror of the reference) ----------
__device__ __forceinline__ void obs_row(float c_x, float c_y, float v_x, float v_y,
                                        float phi, const float* __restrict__ blocks,
                                        float* __restrict__ w /* 7 outputs */)
{
    const float ap = fabsf(phi);
    const float sa = sinf(ap);            // sin(|phi|)
    const float ca = cosf(ap);            // cos(phi) == cos(|phi|)

    // L0/2 = 0.2f, L1/2 = 0.04f (bit-exact halves of 0.4f / 0.08f)
    const float half_x = ca * 0.2f + sa * 0.04f;
    const float half_y = sa * 0.2f + ca * 0.04f;

    float up   = 2.0f - c_y - half_y;     // Y_HI - c_y - half_y
    float down = c_y - 0.0f - half_y;     // c_y - Y_LO - half_y
    float d1 = 10.0f, d2 = 10.0f;         // FAR

    const float lo = c_x - half_x;
    const float hi = c_x + half_x;

#pragma unroll
    for (int i = 0; i < 3; ++i) {
        const float x  = blocks[4 * i + 0];
        const float y  = blocks[4 * i + 1];
        const float hw = blocks[4 * i + 2] * 0.5f;   // w/2 (exact *0.5)
        const float hh = blocks[4 * i + 3] * 0.5f;   // h/2
        const float xl = x - hw;
        const float xr = x + hw;

        const bool x_in = ((xl <= hi) && (xl >= lo)) ||
                          ((xr <= hi) && (xr >= lo));
        if (x_in) {
            up   = y + hh - c_y - half_y;
            down = c_y - (y - hh) - half_y;
        }

        const bool  passed = hi < xl;
        const float dist   = xl - c_x - half_x;
        const float cylo   = c_y - half_y;
        const float cyhi   = c_y + half_y;
        const bool  y1_out = (cylo < y - hh) || (cylo > y + hh);
        const bool  y2_out = (cyhi > y + hh) || (cyhi < y - hh);

        if (passed && y1_out && (d1 == 10.0f)) d1 = dist;
        if (passed && y2_out && (d2 == 10.0f)) d2 = dist;
    }

    w[0] = d1; w[1] = d2; w[2] = up; w[3] = down;
    w[4] = v_x; w[5] = v_y; w[6] = phi;
}

// ---- hot kernel: full 256-row tiles, async DMA staging ---------------------
__global__ __launch_bounds__(ROWS_PER_BLOCK)
void obs_full(const float* __restrict__ state,
              const float* __restrict__ blocks,
              float* __restrict__ out)
{
    __shared__ __align__(16) float lin[ELEMS_PER_BLOCK];
    __shared__ __align__(16) float lout[ELEMS_PER_BLOCK];

    const int tid = threadIdx.x;
    const long long base = (long long)blockIdx.x * ELEMS_PER_BLOCK;

    // Stage input tile: 448 x 16B chunks; threads issue chunk tid and tid+256.
    {
        const int e = tid * 4;
#if HAVE_ASYNC_LDS
        __builtin_amdgcn_global_load_async_to_lds_b128(
            (gv4p)(state + base + e), (lv4p)(lin + e), 0, 0);
        if (e + 1024 < ELEMS_PER_BLOCK)
            __builtin_amdgcn_global_load_async_to_lds_b128(
                (gv4p)(state + base + e + 1024), (lv4p)(lin + e + 1024), 0, 0);
#else
        *(float4*)(lin + e) = *(const float4*)(state + base + e);
        if (e + 1024 < ELEMS_PER_BLOCK)
            *(float4*)(lin + e + 1024) = *(const float4*)(state + base + e + 1024);
#endif
    }
#if HAVE_ASYNC_LDS
    __builtin_amdgcn_s_wait_asynccnt(0);
#endif
    __syncthreads();

    // Per-row math. LDS stride-7 accesses: gcd(7,64)=1 -> conflict-free.
    {
        const float* r = lin + tid * FPR;
        obs_row(r[0], r[1], r[2], r[3], r[4], blocks, lout + tid * FPR);
    }

    __syncthreads();

    // Drain output tile. S_ENDPGM's implicit wait-idle covers completion.
    {
        const int e = tid * 4;
#if HAVE_ASYNC_LDS
        __builtin_amdgcn_global_store_async_from_lds_b128(
            (gv4p)(out + base + e), (lv4p)(lout + e), 0, 0);
        if (e + 1024 < ELEMS_PER_BLOCK)
            __builtin_amdgcn_global_store_async_from_lds_b128(
                (gv4p)(out + base + e + 1024), (lv4p)(lout + e + 1024), 0, 0);
#else
        *(float4*)(out + base + e) = *(const float4*)(lout + e);
        if (e + 1024 < ELEMS_PER_BLOCK)
            *(float4*)(out + base + e + 1024) = *(const float4*)(lout + e + 1024);
#endif
    }
}

// ---- cold kernel: <=255 leftover rows, plain scalar path -------------------
__global__ __launch_bounds__(ROWS_PER_BLOCK)
void obs_tail(const float* __restrict__ state,
              const float* __restrict__ blocks,
              float* __restrict__ out,
              int row0, int rows)
{
    const int row = row0 + (int)threadIdx.x;
    if (row >= rows) return;
    const float* r = state + (long long)row * FPR;
    float w[FPR];
    obs_row(r[0], r[1], r[2], r[3], r[4], blocks, w);
    float* o = out + (long long)row * FPR;
#pragma unroll
    for (int j = 0; j < FPR; ++j) o[j] = w[j];
}

extern "C" void kernel_launch(void* const* d_in, const int* in_sizes, int n_in,
                              void* d_out, int out_size, void* d_ws, size_t ws_size,
                              hipStream_t stream)
{
    const float* state  = (const float*)d_in[0];
    const float* blocks = (const float*)d_in[1];
    float* out = (float*)d_out;

    const int rows  = in_sizes[0] / FPR;        // 4,194,304
    const int nfull = rows / ROWS_PER_BLOCK;    // 16384 (no remainder here)
    const int tail  = rows - nfull * ROWS_PER_BLOCK;

    if (nfull > 0)
        obs_full<<<nfull, ROWS_PER_BLOCK, 0, stream>>>(state, blocks, out);
    if (tail > 0)
        obs_tail<<<1, ROWS_PER_BLOCK, 0, stream>>>(state, blocks, out,
                                                   nfull * ROWS_PER_BLOCK, rows);
}